// UnlearnRecLoss_29858612642065
// MI455X (gfx1250) — compile-verified
//
#include <hip/hip_runtime.h>
#include <hip/hip_bf16.h>
#include <math.h>
#include <stdint.h>

typedef __attribute__((ext_vector_type(2))) float v2f;
typedef __attribute__((ext_vector_type(8))) float v8f;

#define N_PRED 10000
#define N_UNL  4096
#define NEMB   20000
#define EMB_D  128
#define N_REM  8192
#define N_A    8192
#define H_D    128
#define EPSF   1e-8f
#define KT       64
#define A_STRIDE 68   // floats per LDS A row (pad: bank = (4*row+k)%64, conflict-free)
#define M_STRIDE 68   // bytes per LDS mask row (gcd(17,64)=1 -> conflict-free)

__device__ __forceinline__ float softplusf(float x) {
    return fmaxf(x, 0.f) + log1pf(expf(-fabsf(x)));
}
// order-preserving float<->uint map for atomicMax on floats (incl. negatives)
__device__ __forceinline__ unsigned f2ord(float f) {
    unsigned u = __float_as_uint(f);
    return (u & 0x80000000u) ? ~u : (u | 0x80000000u);
}
__device__ __forceinline__ float ord2f(unsigned u) {
    return (u & 0x80000000u) ? __uint_as_float(u & 0x7fffffffu)
                             : __uint_as_float(~u);
}
__device__ __forceinline__ float wave_sum(float v) {
    for (int off = 16; off > 0; off >>= 1) v += __shfl_xor(v, off, 32);
    return v;
}
// low 32 bits of a generic pointer to an LDS object == wave-relative LDS byte
// address (ISA 10.2: LDS aperture)
__device__ __forceinline__ unsigned lds_addr(const void* p) {
    return (unsigned)(uintptr_t)p;
}

// ---- init: zero accumulators, set segment maxes to ord(-inf) ----
__global__ void k_init(unsigned* m_o, unsigned* m_n, float* s_o, float* s_n,
                       float* cnt, float* acc) {
    int i = blockIdx.x * blockDim.x + threadIdx.x;
    if (i < NEMB) {
        m_o[i] = 0x007FFFFFu;  // f2ord(-inf)
        m_n[i] = 0x007FFFFFu;
        s_o[i] = 0.f; s_n[i] = 0.f; cnt[i] = 0.f;
    }
    if (i < 8) acc[i] = 0.f;
}

// ---- L_u: -mean(log_sigmoid(-s)) = mean(softplus(s)) ----
__global__ void k_lu(const float* __restrict__ pred, const int* __restrict__ edges,
                     float* acc) {
    int e = blockIdx.x * blockDim.x + threadIdx.x;
    float v = 0.f;
    if (e < N_UNL) {
        long u = edges[2 * e], w = edges[2 * e + 1];
        v = softplusf(pred[u * (long)N_PRED + w]);
    }
    v = wave_sum(v);
    if ((threadIdx.x & 31) == 0) atomicAdd(&acc[0], v);
}

// ---- per-edge sims (both embeddings), segment max + counts ----
__global__ void k_sims(const float* __restrict__ Eo, const float* __restrict__ En,
                       const int* __restrict__ edges,
                       float* sims_o, float* sims_n,
                       unsigned* m_o, unsigned* m_n, float* cnt) {
    int wave = (blockIdx.x * blockDim.x + threadIdx.x) >> 5;
    int lane = threadIdx.x & 31;
    if (wave >= N_REM) return;
    long u = edges[2 * wave], v = edges[2 * wave + 1];
    const float* au = Eo + u * EMB_D; const float* av = Eo + v * EMB_D;
    const float* bu = En + u * EMB_D; const float* bv = En + v * EMB_D;
    float so = 0.f, sn = 0.f;
    for (int k = lane; k < EMB_D; k += 32) { so += au[k] * av[k]; sn += bu[k] * bv[k]; }
    so = wave_sum(so); sn = wave_sum(sn);   // TAU = 1
    if (lane == 0) {
        sims_o[wave] = so; sims_n[wave] = sn;
        atomicMax(&m_o[u], f2ord(so));
        atomicMax(&m_n[u], f2ord(sn));
        atomicAdd(&cnt[u], 1.f);
    }
}

// ---- segment sum of exp(sims - m_safe[u]) ----
__global__ void k_segsum(const int* __restrict__ edges,
                         const float* __restrict__ sims_o, const float* __restrict__ sims_n,
                         const unsigned* __restrict__ m_o, const unsigned* __restrict__ m_n,
                         float* s_o, float* s_n) {
    int e = blockIdx.x * blockDim.x + threadIdx.x;
    if (e >= N_REM) return;
    int u = edges[2 * e];
    float mo = ord2f(m_o[u]); mo = isfinite(mo) ? mo : 0.f;
    float mn = ord2f(m_n[u]); mn = isfinite(mn) ? mn : 0.f;
    atomicAdd(&s_o[u], expf(sims_o[e] - mo));
    atomicAdd(&s_n[u], expf(sims_n[e] - mn));
}

// ---- L_p numerator/denominator ----
__global__ void k_lp(const int* __restrict__ edges,
                     const float* __restrict__ sims_o, const float* __restrict__ sims_n,
                     const unsigned* __restrict__ m_o, const unsigned* __restrict__ m_n,
                     const float* __restrict__ s_o, const float* __restrict__ s_n,
                     const float* __restrict__ cnt, float* acc) {
    int e = blockIdx.x * blockDim.x + threadIdx.x;
    if (e >= N_REM) return;
    int u = edges[2 * e];
    float mo = ord2f(m_o[u]); mo = isfinite(mo) ? mo : 0.f;
    float mn = ord2f(m_n[u]); mn = isfinite(mn) ? mn : 0.f;
    float lse_o = logf(s_o[u] + 1e-30f) + mo;
    float lse_n = logf(s_n[u] + 1e-30f) + mn;
    float lpo = logf(expf(sims_o[e] - lse_o) + EPSF);
    float lpn = logf(expf(sims_n[e] - lse_n) + EPSF);
    float mk = (cnt[u] > 1.f) ? 1.f : 0.f;
    float d = lpn - lpo;
    atomicAdd(&acc[1], mk * d * d);
    atomicAdd(&acc[2], mk);
}

// ---- dinv[i] = rsqrt(sum_j A*mask + eps) : one block per row ----
__global__ void __launch_bounds__(256) k_dinv(const float* __restrict__ A,
                                              const unsigned char* __restrict__ mask,
                                              float* dinv) {
    long row = blockIdx.x;
    const float* ar = A + row * N_A;
    const unsigned char* mr = mask + row * N_A;
    float s = 0.f;
    for (int j = threadIdx.x; j < N_A; j += 256) s += ar[j] * (float)mr[j];
    s = wave_sum(s);
    __shared__ float red[8];
    if ((threadIdx.x & 31) == 0) red[threadIdx.x >> 5] = s;
    __syncthreads();
    if (threadIdx.x == 0) {
        float t = 0.f;
        for (int i = 0; i < 8; i++) t += red[i];
        dinv[row] = rsqrtf(t + EPSF);
    }
}

// ---- Hs[j,:] = dinv[j] * H[j,:] ----
__global__ void k_hs(const float* __restrict__ H, const float* __restrict__ dinv,
                     float* Hs) {
    int i = blockIdx.x * blockDim.x + threadIdx.x;
    if (i < N_A * H_D) Hs[i] = dinv[i >> 7] * H[i];
}

// ---- Hp = dinv ⊙ ((A⊙mask) @ Hs), fp32 WMMA 16x16x4 ----
// block: 256 thr = 8 waves; block owns 16 M-rows; wave w owns N-cols [16w,16w+16)
// A/mask tiles double-buffered in LDS via async global->LDS copies (ASYNCcnt).
__global__ void __launch_bounds__(256) k_gemm(const float* __restrict__ A,
                                              const unsigned char* __restrict__ mask,
                                              const float* __restrict__ Hs,
                                              const float* __restrict__ dinv,
                                              float* __restrict__ Hp) {
    __shared__ __align__(16) float         As[2][16 * A_STRIDE];
    __shared__ __align__(16) unsigned char Ms[2][16 * M_STRIDE];

    const int i0    = blockIdx.x * 16;
    const int wv    = threadIdx.x >> 5;
    const int lane  = threadIdx.x & 31;
    const int col   = lane & 15;       // A: M row / B,D: N col
    const int khalf = lane >> 4;       // selects K pair {0,1} vs {2,3}
    const int n0    = wv * 16;

    // staging coords: thread t copies 16B of A (4 floats) + 4B of mask
    const int sr = threadIdx.x >> 4;   // 0..15 tile row
    const int sc = threadIdx.x & 15;   // 0..15 16B chunk
    const float*         ga_base = A    + (long)(i0 + sr) * N_A + sc * 4;
    const unsigned char* gm_base = mask + (long)(i0 + sr) * N_A + sc * 4;
    unsigned ldsA0 = lds_addr(&As[0][sr * A_STRIDE + sc * 4]);
    unsigned ldsA1 = lds_addr(&As[1][sr * A_STRIDE + sc * 4]);
    unsigned ldsM0 = lds_addr(&Ms[0][sr * M_STRIDE + sc * 4]);
    unsigned ldsM1 = lds_addr(&Ms[1][sr * M_STRIDE + sc * 4]);

    // prologue: async-stage tile 0 into buffer 0
    asm volatile("global_load_async_to_lds_b128 %0, %1, off"
                 :: "v"(ldsA0), "v"(ga_base) : "memory");
    asm volatile("global_load_async_to_lds_b32 %0, %1, off"
                 :: "v"(ldsM0), "v"(gm_base) : "memory");

    v8f acc = {};
    const int a_off = col * A_STRIDE;
    const int m_off = col * M_STRIDE;

    for (int k0 = 0; k0 < N_A; k0 += KT) {
        const int b = (k0 / KT) & 1;
        const int kn = k0 + KT;
        if (kn < N_A) {  // async-stage next tile into the other buffer
            asm volatile("global_load_async_to_lds_b128 %0, %1, off"
                         :: "v"(b ? ldsA0 : ldsA1), "v"(ga_base + kn) : "memory");
            asm volatile("global_load_async_to_lds_b32 %0, %1, off"
                         :: "v"(b ? ldsM0 : ldsM1), "v"(gm_base + kn) : "memory");
            // <=2 outstanding: the 2 just-issued; current tile's copies are done
            asm volatile("s_wait_asynccnt 0x2" ::: "memory");
        } else {
            asm volatile("s_wait_asynccnt 0x0" ::: "memory");
        }
        __syncthreads();   // publish tile b to all waves

        const float*         as = &As[b][a_off];
        const unsigned char* ms = &Ms[b][m_off];
        for (int kk = 0; kk < KT; kk += 4) {
            const int ka = kk + khalf * 2;
            // unconditional 8B LDS load + 2B mask load; mask applied as
            // arithmetic (bytes are 0/1) -> no predicated loads / EXEC writes
            v2f av = *(const v2f*)(as + ka);
            unsigned m2 = *(const unsigned short*)(ms + ka);
            v2f a, bb;
            a.x = av.x * (float)(m2 & 0xffu);
            a.y = av.y * (float)(m2 >> 8);
            long bidx = (long)(k0 + ka) * H_D + n0 + col;
            bb.x = Hs[bidx];
            bb.y = Hs[bidx + H_D];
            acc = __builtin_amdgcn_wmma_f32_16x16x4_f32(
                false, a, false, bb, (short)0, acc, false, false);
        }
        __syncthreads();   // all waves done reading buf b before it is rewritten
    }
    // D layout: VGPR r -> row r (lanes 0-15) / row r+8 (lanes 16-31), N = lane&15
    const int rbase = khalf * 8;
    for (int r = 0; r < 8; r++) {
        int row = i0 + rbase + r;
        Hp[(long)row * H_D + n0 + col] = dinv[row] * acc[r];
    }
}

// ---- cosine + -log_sigmoid, one wave per row ----
__global__ void k_cos(const float* __restrict__ H, const float* __restrict__ Hp,
                      float* acc) {
    int wave = (blockIdx.x * blockDim.x + threadIdx.x) >> 5;
    int lane = threadIdx.x & 31;
    if (wave >= N_A) return;
    const float* h  = H  + (long)wave * H_D;
    const float* hp = Hp + (long)wave * H_D;
    float num = 0.f, nh = 0.f, np = 0.f;
    for (int k = lane; k < H_D; k += 32) {
        float a = h[k], b = hp[k];
        num += a * b; nh += a * a; np += b * b;
    }
    num = wave_sum(num); nh = wave_sum(nh); np = wave_sum(np);
    if (lane == 0) {
        float den = fmaxf(sqrtf(nh) * sqrtf(np), EPSF);
        atomicAdd(&acc[3], softplusf(-(num / den)));  // -log_sigmoid(cos), T=1
    }
}

// ---- combine ----
__global__ void k_final(const float* __restrict__ acc, float* out) {
    if (blockIdx.x == 0 && threadIdx.x == 0) {
        float Lu = acc[0] / (float)N_UNL;
        float Lp = acc[1] / fmaxf(acc[2], 1.f);
        float Lc = acc[3] / (float)N_A;
        out[0] = Lu + Lp + 0.01f * Lc;
        out[1] = Lu; out[2] = Lp; out[3] = Lc;
    }
}

extern "C" void kernel_launch(void* const* d_in, const int* in_sizes, int n_in,
                              void* d_out, int out_size, void* d_ws, size_t ws_size,
                              hipStream_t stream) {
    const float* pred = (const float*)d_in[0];
    const float* Eo   = (const float*)d_in[1];
    const float* En   = (const float*)d_in[2];
    const float* A    = (const float*)d_in[3];
    const float* H    = (const float*)d_in[4];
    const int* unl    = (const int*)d_in[5];
    const int* rem    = (const int*)d_in[6];
    const unsigned char* dmask = (const unsigned char*)d_in[7];
    float* out = (float*)d_out;

    // workspace layout
    float* ws   = (float*)d_ws;
    float* Hs   = ws;                          // 8192*128
    float* Hp   = Hs + (long)N_A * H_D;        // 8192*128
    float* dinv = Hp + (long)N_A * H_D;        // 8192
    unsigned* m_o = (unsigned*)(dinv + N_A);   // 20000
    unsigned* m_n = m_o + NEMB;                // 20000
    float* s_o  = (float*)(m_n + NEMB);        // 20000
    float* s_n  = s_o + NEMB;                  // 20000
    float* cnt  = s_n + NEMB;                  // 20000
    float* sims_o = cnt + NEMB;                // 8192
    float* sims_n = sims_o + N_REM;            // 8192
    float* acc  = sims_n + N_REM;              // 8

    k_init  <<<(NEMB + 255) / 256, 256, 0, stream>>>(m_o, m_n, s_o, s_n, cnt, acc);
    k_lu    <<<(N_UNL + 255) / 256, 256, 0, stream>>>(pred, unl, acc);
    k_sims  <<<(N_REM * 32) / 256, 256, 0, stream>>>(Eo, En, rem, sims_o, sims_n, m_o, m_n, cnt);
    k_segsum<<<N_REM / 256, 256, 0, stream>>>(rem, sims_o, sims_n, m_o, m_n, s_o, s_n);
    k_lp    <<<N_REM / 256, 256, 0, stream>>>(rem, sims_o, sims_n, m_o, m_n, s_o, s_n, cnt, acc);
    k_dinv  <<<N_A, 256, 0, stream>>>(A, dmask, dinv);
    k_hs    <<<(N_A * H_D) / 256, 256, 0, stream>>>(H, dinv, Hs);
    k_gemm  <<<N_A / 16, 256, 0, stream>>>(A, dmask, Hs, dinv, Hp);
    k_cos   <<<(N_A * 32) / 256, 256, 0, stream>>>(H, Hp, acc);
    k_final <<<1, 32, 0, stream>>>(acc, out);
}